// yolov1_loss_71451075936292
// MI455X (gfx1250) — compile-verified
//
#include <hip/hip_runtime.h>

typedef float v2f __attribute__((ext_vector_type(2)));
typedef float v8f __attribute__((ext_vector_type(8)));

#define THREADS 256
#define HW 784            // 28*28
#define QPP 196           // float4-quads per image plane (784/4)
#define NBLOCKS 392       // 512*784/4/256

// ---- exact 32-lane wave sum via V_WMMA_F32_16X16X4_F32 (ones-matrix trick) ----
// A (16x4, documented layout): v0 -> A[m][0]=x[m] (lanes 0-15), A[m][2]=x[m+16]
// (lanes 16-31); v1 = 0. B = all-ones (layout-invariant). D[m][n] = x[m]+x[m+16].
// Sum of the 8 D VGPRs per lane + shfl_xor(16) == exact sum of all 32 lanes.
__device__ __forceinline__ float wave_sum32(float x) {
  v2f a; a[0] = x;    a[1] = 0.0f;
  v2f b; b[0] = 1.0f; b[1] = 1.0f;
  v8f c = {0.f, 0.f, 0.f, 0.f, 0.f, 0.f, 0.f, 0.f};
  v8f d = __builtin_amdgcn_wmma_f32_16x16x4_f32(
      /*neg_a=*/false, a, /*neg_b=*/false, b,
      /*c_mod=*/(short)0, c, /*reuse_a=*/false, /*reuse_b=*/false);
  float s = ((d[0] + d[1]) + (d[2] + d[3])) + ((d[4] + d[5]) + (d[6] + d[7]));
  s += __shfl_xor(s, 16, 32);
  return s;  // full wave sum broadcast to all 32 lanes
}

__device__ __forceinline__ float4 ld4(const float* __restrict__ p, int chan, int hw) {
  return *reinterpret_cast<const float4*>(p + (size_t)chan * HW + hw);
}

__device__ __forceinline__ float fget(const float4& v, int j) {
  switch (j) { case 0: return v.x; case 1: return v.y; case 2: return v.z; default: return v.w; }
}

__global__ __launch_bounds__(THREADS) void yolo_part_kernel(
    const float* __restrict__ pred_cls,
    const float* __restrict__ pred_resp,
    const float* __restrict__ pred_bb,
    const float* __restrict__ label_cls,
    const float* __restrict__ label_resp,
    const float* __restrict__ label_bb,
    float* __restrict__ block_part) {
  const int q    = blockIdx.x * THREADS + threadIdx.x;  // quad id, 0..100351
  const int bimg = q / QPP;
  const int hw0  = (q - bimg * QPP) * 4;                // 16B-aligned, same image

  // responses (Bb = 2)
  float4 lr0 = ld4(label_resp, bimg * 2 + 0, hw0);
  float4 lr1 = ld4(label_resp, bimg * 2 + 1, hw0);
  float4 pr0 = ld4(pred_resp,  bimg * 2 + 0, hw0);
  float4 pr1 = ld4(pred_resp,  bimg * 2 + 1, hw0);

  // bbox offsets: channel order box0:(ox,oy,sw,sh), box1:(ox,oy,sw,sh)
  float4 t[8], p[8];
#pragma unroll
  for (int c = 0; c < 8; ++c) {
    t[c] = ld4(label_bb, bimg * 8 + c, hw0);
    p[c] = ld4(pred_bb,  bimg * 8 + c, hw0);
  }

  float acc_obj = 0.f, acc_off = 0.f;
  float maskj[4];

#pragma unroll
  for (int j = 0; j < 4; ++j) {
    const int cell = hw0 + j;
    const int hy = cell / 28;
    const int wx = cell - hy * 28;
    const float cx = (float)wx, cy = (float)hy;

    float iou[2], area_t[2];
#pragma unroll
    for (int bb = 0; bb < 2; ++bb) {
      const int o = bb * 4;
      const float tox = fget(t[o + 0], j), toy = fget(t[o + 1], j);
      const float tsw = fget(t[o + 2], j), tsh = fget(t[o + 3], j);
      const float pox = fget(p[o + 0], j), poy = fget(p[o + 1], j);
      const float psw = fget(p[o + 2], j), psh = fget(p[o + 3], j);
      const float tw = tsw * tsw, th = tsh * tsh;
      const float pw = psw * psw, ph = psh * psh;
      const float tx1 = (cx - tox) - 0.5f * tw, ty1 = (cy - toy) - 0.5f * th;
      const float px1 = (cx - pox) - 0.5f * pw, py1 = (cy - poy) - 0.5f * ph;
      const float lx = fmaxf(tx1, px1), ly = fmaxf(ty1, py1);
      const float rx = fminf(tx1 + tw, px1 + pw), ry = fminf(ty1 + th, py1 + ph);
      const float iw = fmaxf(rx - lx, 0.f), ih = fmaxf(ry - ly, 0.f);
      const float inter = iw * ih;
      area_t[bb] = tw * th;
      iou[bb] = inter / (tw * th + pw * ph - inter);
    }
    // jnp.argmax: first max on ties -> best=1 only if strictly greater
    const int best = (iou[1] > iou[0]) ? 1 : 0;
    const float m = (fget(lr0, j) + fget(lr1, j) >= 0.5f) ? 1.f : 0.f;
    maskj[j] = m;

    const float presp = best ? fget(pr1, j) : fget(pr0, j);
    const float dobj = presp - iou[best];
    acc_obj += m * dobj * dobj;

    float offs = 0.f;
    const int ob = best * 4;
#pragma unroll
    for (int k = 0; k < 4; ++k) {
      const float dv = fget(t[ob + k], j) - fget(p[ob + k], j);
      offs += dv * dv;
    }
    acc_off += m * offs;

    // negative-response term (folded into the same output component as pos-response)
    {
      const float l0 = fget(lr0, j), l1 = fget(lr1, j);
      const float n0 = (l0 < 1.0f) ? 1.f : 0.f;
      const float n1 = (l1 < 1.0f) ? 1.f : 0.f;
      const float d0 = fget(pr0, j) - l0;
      const float d1 = fget(pr1, j) - l1;
      acc_obj += n0 * d0 * d0 + n1 * d1 * d1;
    }
    (void)area_t;
  }

  // classification loss: stream 20 channels of pred/label cls
  float csum0 = 0.f, csum1 = 0.f, csum2 = 0.f, csum3 = 0.f;
  for (int c = 0; c < 20; ++c) {
    const float4 pc = ld4(pred_cls,  bimg * 20 + c, hw0);
    const float4 lc = ld4(label_cls, bimg * 20 + c, hw0);
    float d;
    d = pc.x - lc.x; csum0 += d * d;
    d = pc.y - lc.y; csum1 += d * d;
    d = pc.z - lc.z; csum2 += d * d;
    d = pc.w - lc.w; csum3 += d * d;
  }
  const float acc_cls = maskj[0] * csum0 + maskj[1] * csum1 +
                        maskj[2] * csum2 + maskj[3] * csum3;

  // ---- block reduction: WMMA wave sums, then LDS across the 8 waves ----
  const float s0 = wave_sum32(acc_obj);
  const float s1 = wave_sum32(acc_cls);
  const float s2 = wave_sum32(acc_off);

  __shared__ float red[8][3];
  const int wid = threadIdx.x >> 5;
  if ((threadIdx.x & 31) == 0) { red[wid][0] = s0; red[wid][1] = s1; red[wid][2] = s2; }
  __syncthreads();
  if (threadIdx.x == 0) {
    float r0 = 0.f, r1 = 0.f, r2 = 0.f;
#pragma unroll
    for (int i = 0; i < 8; ++i) { r0 += red[i][0]; r1 += red[i][1]; r2 += red[i][2]; }
    float* o3 = block_part + (size_t)blockIdx.x * 3;
    o3[0] = r0; o3[1] = r1; o3[2] = r2;
  }
}

__global__ __launch_bounds__(THREADS) void yolo_final_kernel(
    const float* __restrict__ part, int nparts, float inv_bs,
    float* __restrict__ out) {
  float a0 = 0.f, a1 = 0.f, a2 = 0.f;
  for (int i = threadIdx.x; i < nparts; i += THREADS) {
    a0 += part[i * 3 + 0];
    a1 += part[i * 3 + 1];
    a2 += part[i * 3 + 2];
  }
  const float s0 = wave_sum32(a0);
  const float s1 = wave_sum32(a1);
  const float s2 = wave_sum32(a2);

  __shared__ float red[8][3];
  const int wid = threadIdx.x >> 5;
  if ((threadIdx.x & 31) == 0) { red[wid][0] = s0; red[wid][1] = s1; red[wid][2] = s2; }
  __syncthreads();
  if (threadIdx.x == 0) {
    float r0 = 0.f, r1 = 0.f, r2 = 0.f;
#pragma unroll
    for (int i = 0; i < 8; ++i) { r0 += red[i][0]; r1 += red[i][1]; r2 += red[i][2]; }
    out[0] = r0 * inv_bs;  // l_obj = neg_response + pos_response
    out[1] = r1 * inv_bs;  // l_cls
    out[2] = r2 * inv_bs;  // l_offset
  }
}

extern "C" void kernel_launch(void* const* d_in, const int* in_sizes, int n_in,
                              void* d_out, int out_size, void* d_ws, size_t ws_size,
                              hipStream_t stream) {
  const float* pred_cls   = (const float*)d_in[0];
  const float* pred_resp  = (const float*)d_in[1];
  const float* pred_bb    = (const float*)d_in[2];
  const float* label_cls  = (const float*)d_in[3];
  const float* label_resp = (const float*)d_in[4];
  const float* label_bb   = (const float*)d_in[5];
  float* part = (float*)d_ws;   // NBLOCKS*3 floats of scratch
  float* out  = (float*)d_out;  // 3 floats

  yolo_part_kernel<<<NBLOCKS, THREADS, 0, stream>>>(
      pred_cls, pred_resp, pred_bb, label_cls, label_resp, label_bb, part);
  yolo_final_kernel<<<1, THREADS, 0, stream>>>(part, NBLOCKS, 1.0f / 512.0f, out);
}